// AggregateModule_21526376087641
// MI455X (gfx1250) — compile-verified
//
#include <hip/hip_runtime.h>
#include <math.h>

#define N_NODES 131072
#define E_EDGES 1048576
#define NUM_DSTN 65536
#define IN_DIM 256
#define HEADS 4
#define OUT_D 32
#define FEAT (HEADS * OUT_D)   // 128
#define NEG_SLOPE 0.2f

typedef __attribute__((ext_vector_type(2))) float v2f;
typedef __attribute__((ext_vector_type(8))) float v8f;

// ---------------------------------------------------------------- init
__global__ void init_kernel(float* __restrict__ out,
                            float* __restrict__ emax,
                            float* __restrict__ esum) {
    size_t i = (size_t)blockIdx.x * blockDim.x + threadIdx.x;
    if (i < (size_t)NUM_DSTN * FEAT) out[i] = 0.0f;
    if (i < (size_t)NUM_DSTN * HEADS) {
        ((unsigned int*)emax)[i] = 0xFF800000u;  // -inf
        esum[i] = 0.0f;
    }
}

// ---------------------------------------------------------------- GEMM: h = feats @ W
// One wave computes a 16-row x 128-col strip of h using V_WMMA_F32_16X16X4_F32.
// A frag (16x4 f32): lanes 0-15 hold M=lo; v.x = K=2*hi, v.y = K=2*hi+1.
// B frag (4x16 f32): lanes 0-15 = K row 2*hi (N=lo), .y = row 2*hi+1.
// D (16x16 f32): VGPR i -> row i + 8*hi, col = lo.
__global__ __launch_bounds__(256) void gemm_kernel(const float* __restrict__ feats,
                                                   const float* __restrict__ W,
                                                   float* __restrict__ h) {
    const int wave = blockIdx.x * (blockDim.x >> 5) + (threadIdx.x >> 5);
    const int lane = threadIdx.x & 31;
    const int lo = lane & 15;
    const int hi = lane >> 4;
    const int rowBase = wave * 16;
    const float* __restrict__ arow = feats + (size_t)(rowBase + lo) * IN_DIM;

    v8f acc[8];
#pragma unroll
    for (int t = 0; t < 8; ++t) acc[t] = (v8f){0.f, 0.f, 0.f, 0.f, 0.f, 0.f, 0.f, 0.f};

    for (int k = 0; k < IN_DIM; k += 4) {
        const int ka = k + 2 * hi;
        v2f a;
        a.x = arow[ka];
        a.y = arow[ka + 1];
#pragma unroll
        for (int t = 0; t < 8; ++t) {
            const int col = t * 16 + lo;
            v2f b;
            b.x = W[(size_t)ka * FEAT + col];
            b.y = W[(size_t)(ka + 1) * FEAT + col];
            acc[t] = __builtin_amdgcn_wmma_f32_16x16x4_f32(
                false, a, false, b, (short)0, acc[t], false, false);
        }
    }

#pragma unroll
    for (int t = 0; t < 8; ++t)
#pragma unroll
        for (int i = 0; i < 8; ++i)
            h[(size_t)(rowBase + i + 8 * hi) * FEAT + t * 16 + lo] = acc[t][i];
}

// ---------------------------------------------------------------- attention logits
__global__ void attn_logits_kernel(const float* __restrict__ h,
                                   const float* __restrict__ attn_l,
                                   const float* __restrict__ attn_r,
                                   float* __restrict__ el, float* __restrict__ er) {
    int i = blockIdx.x * blockDim.x + threadIdx.x;   // (node * HEADS + head)
    if (i >= N_NODES * HEADS) return;
    const int n = i >> 2;
    const int hh = i & 3;
    const float* __restrict__ hp = h + (size_t)n * FEAT + hh * OUT_D;
    const float* __restrict__ al = attn_l + hh * OUT_D;
    const float* __restrict__ ar = attn_r + hh * OUT_D;
    float sl = 0.f, sr = 0.f;
#pragma unroll
    for (int d = 0; d < OUT_D; ++d) {
        float v = hp[d];
        sl += v * al[d];
        sr += v * ar[d];
    }
    el[i] = sl;
    er[i] = sr;
}

// ---------------------------------------------------------------- float atomic max
__device__ __forceinline__ void atomicMaxF(float* addr, float val) {
    if (val >= 0.0f)
        atomicMax((int*)addr, __float_as_int(val));
    else
        atomicMin((unsigned int*)addr, __float_as_uint(val));
}

// ---------------------------------------------------------------- edge pass 1: e, segment max
__global__ void edge_max_kernel(const int* __restrict__ src, const int* __restrict__ dst,
                                const float* __restrict__ el, const float* __restrict__ er,
                                float* __restrict__ ebuf, float* __restrict__ emax) {
    int e = blockIdx.x * blockDim.x + threadIdx.x;
    if (e >= E_EDGES) return;
    const int s = src[e];
    const int d = dst[e];
    const float4 l4 = *(const float4*)(el + (size_t)s * HEADS);
    const float4 r4 = *(const float4*)(er + (size_t)d * HEADS);
    float vals[4] = {l4.x + r4.x, l4.y + r4.y, l4.z + r4.z, l4.w + r4.w};
#pragma unroll
    for (int hh = 0; hh < 4; ++hh) {
        float v = vals[hh];
        v = (v >= 0.f) ? v : NEG_SLOPE * v;
        ebuf[(size_t)e * HEADS + hh] = v;
        atomicMaxF(&emax[(size_t)d * HEADS + hh], v);
    }
}

// ---------------------------------------------------------------- edge pass 2: exp, segment sum
__global__ void edge_exp_kernel(const int* __restrict__ dst,
                                float* __restrict__ ebuf,
                                const float* __restrict__ emax,
                                float* __restrict__ esum) {
    int e = blockIdx.x * blockDim.x + threadIdx.x;
    if (e >= E_EDGES) return;
    const int d = dst[e];
#pragma unroll
    for (int hh = 0; hh < 4; ++hh) {
        float v = expf(ebuf[(size_t)e * HEADS + hh] - emax[(size_t)d * HEADS + hh]);
        ebuf[(size_t)e * HEADS + hh] = v;
        atomicAdd(&esum[(size_t)d * HEADS + hh], v);
    }
}

// ---------------------------------------------------------------- edge pass 3: scatter msg
// One block of 128 threads per edge; thread t handles feature t (head = t>>5).
__global__ __launch_bounds__(128) void scatter_kernel(const int* __restrict__ src,
                                                      const int* __restrict__ dst,
                                                      const float* __restrict__ h,
                                                      const float* __restrict__ ebuf,
                                                      const float* __restrict__ esum,
                                                      float* __restrict__ out) {
    const int e = blockIdx.x;
    const int t = threadIdx.x;
    const int s = src[e];
    const int d = dst[e];
    const int hh = t >> 5;
    const float alpha = ebuf[(size_t)e * HEADS + hh] / esum[(size_t)d * HEADS + hh];
    const float v = h[(size_t)s * FEAT + t] * alpha;
    atomicAdd(&out[(size_t)d * FEAT + t], v);
}

// ---------------------------------------------------------------- finalize: elu in place + att_sc
__global__ void finalize_kernel(float* __restrict__ out) {
    size_t i = (size_t)blockIdx.x * blockDim.x + threadIdx.x;
    if (i < (size_t)NUM_DSTN * FEAT) {
        float x = out[i];
        out[i] = (x > 0.f) ? x : (expf(x) - 1.0f);
    }
    if (i == 0) {
        out[(size_t)NUM_DSTN * FEAT + 0] = 1.0f;   // att_sc
        out[(size_t)NUM_DSTN * FEAT + 1] = 1.0f;
    }
}

// ---------------------------------------------------------------- launch
extern "C" void kernel_launch(void* const* d_in, const int* in_sizes, int n_in,
                              void* d_out, int out_size, void* d_ws, size_t ws_size,
                              hipStream_t stream) {
    const float* feats  = (const float*)d_in[0];
    const int*   src    = (const int*)d_in[1];
    const int*   dst    = (const int*)d_in[2];
    const float* W      = (const float*)d_in[3];
    const float* attn_l = (const float*)d_in[4];
    const float* attn_r = (const float*)d_in[5];
    float* out = (float*)d_out;

    // workspace layout (floats)
    float* ws   = (float*)d_ws;
    float* hbuf = ws;                                   // N * 128
    float* el   = hbuf + (size_t)N_NODES * FEAT;        // N * 4
    float* er   = el + (size_t)N_NODES * HEADS;         // N * 4
    float* ebuf = er + (size_t)N_NODES * HEADS;         // E * 4
    float* emax = ebuf + (size_t)E_EDGES * HEADS;       // NUM_DST * 4
    float* esum = emax + (size_t)NUM_DSTN * HEADS;      // NUM_DST * 4

    // 0) init accumulators / softmax state
    {
        size_t n = (size_t)NUM_DSTN * FEAT;
        init_kernel<<<(unsigned)((n + 255) / 256), 256, 0, stream>>>(out, emax, esum);
    }
    // 1) h = feats @ W  (fp32 WMMA, 16-row strip per wave)
    {
        int rowTiles = N_NODES / 16;            // 8192 waves
        int blocks = rowTiles / 8;              // 8 waves per 256-thread block
        gemm_kernel<<<blocks, 256, 0, stream>>>(feats, W, hbuf);
    }
    // 2) el/er
    {
        int n = N_NODES * HEADS;
        attn_logits_kernel<<<(n + 255) / 256, 256, 0, stream>>>(hbuf, attn_l, attn_r, el, er);
    }
    // 3) e = leaky_relu(el[src]+er[dst]); segment max
    edge_max_kernel<<<(E_EDGES + 255) / 256, 256, 0, stream>>>(src, dst, el, er, ebuf, emax);
    // 4) ee = exp(e - max); segment sum
    edge_exp_kernel<<<(E_EDGES + 255) / 256, 256, 0, stream>>>(dst, ebuf, emax, esum);
    // 5) scatter-add alpha * h[src] into out
    scatter_kernel<<<E_EDGES, 128, 0, stream>>>(src, dst, hbuf, ebuf, esum, out);
    // 6) elu in place + att_sc ones
    {
        size_t n = (size_t)NUM_DSTN * FEAT;
        finalize_kernel<<<(unsigned)((n + 255) / 256), 256, 0, stream>>>(out);
    }
}